// MyHMM_14353780703777
// MI455X (gfx1250) — compile-verified
//
#include <hip/hip_runtime.h>
#include <hip/hip_bf16.h>
#include <math.h>

#define BB 128
#define TT 2050
#define CC 64
#define XD 6
#define LOG2PI_F 1.8378770664093453f

typedef __attribute__((ext_vector_type(2))) float v2f;
typedef __attribute__((ext_vector_type(8))) float v8f;

// ---------------------------------------------------------------------------
// K0: normalize log_A / log_pi, build A=exp(logA), and the 12x64 emission
//     coefficient matrix W:  logp[r,c] = [x0^2..x3^2, x0..x3, 1, 0,0,0] . W[:,c]
// ---------------------------------------------------------------------------
__global__ void hmm_prep(const float* __restrict__ log_A,
                         const float* __restrict__ log_pi,
                         const float* __restrict__ means,
                         const float* __restrict__ stds,
                         float* __restrict__ Aout,      // [64][64] (from j, to c)
                         float* __restrict__ logAout,   // [64][64]
                         float* __restrict__ logpiout,  // [64]
                         float* __restrict__ W)         // [12][64]
{
    int c = threadIdx.x;  // 0..63, used as row index for log_A softmax
    // log_softmax over axis=1 of log_A row c
    float m = -INFINITY;
    for (int j = 0; j < CC; ++j) m = fmaxf(m, log_A[c * CC + j]);
    float s = 0.f;
    for (int j = 0; j < CC; ++j) s += expf(log_A[c * CC + j] - m);
    float lse = m + logf(s);
    for (int j = 0; j < CC; ++j) {
        float v = log_A[c * CC + j] - lse;
        logAout[c * CC + j] = v;
        Aout[c * CC + j]    = expf(v);
    }
    // log_softmax of log_pi (redundantly per thread; trivial)
    float mp = -INFINITY;
    for (int j = 0; j < CC; ++j) mp = fmaxf(mp, log_pi[j]);
    float sp = 0.f;
    for (int j = 0; j < CC; ++j) sp += expf(log_pi[j] - mp);
    logpiout[c] = log_pi[c] - (mp + logf(sp));
    // emission coefficients for state c
    float cc = 0.f;
    for (int d = 0; d < 4; ++d) {
        float mu  = means[c * XD + d];
        float sig = fmaxf(stds[c * XD + d], 0.f) + 0.1f;
        float inv2 = 1.f / (sig * sig);
        W[d * CC + c]       = -0.5f * inv2;   // coeff of x^2
        W[(4 + d) * CC + c] = mu * inv2;      // coeff of x
        cc += -0.5f * mu * mu * inv2 - logf(sig) - 0.5f * LOG2PI_F;
    }
    W[8 * CC + c]  = cc;                      // coeff of 1
    W[9 * CC + c]  = 0.f;
    W[10 * CC + c] = 0.f;
    W[11 * CC + c] = 0.f;
}

// ---------------------------------------------------------------------------
// K1: emission GEMM via V_WMMA_F32_16X16X4_F32.
//     logp(262400 x 64) = F(262400 x 12) @ W(12 x 64).
//     One block = one M-tile of 16 rows; 4 waves cover N=64 (16 each);
//     3 wmma per wave (K = 12 in steps of 4).
// ---------------------------------------------------------------------------
__global__ void __launch_bounds__(128) hmm_emission_wmma(
    const float* __restrict__ x,     // [B*T][6]
    const float* __restrict__ W,     // [12][64]
    float* __restrict__ logp)        // [B*T][64]
{
    __shared__ float feat[16][12];
    __shared__ float Wl[12][64];
    int tid = threadIdx.x;           // 0..127
    long tile = blockIdx.x;          // 0..16399

    for (int i = tid; i < 12 * 64; i += 128) (&Wl[0][0])[i] = W[i];
    if (tid < 16) {
        long r = tile * 16 + tid;
        const float* xr = x + r * XD;
        float x0 = xr[0], x1 = xr[1], x2 = xr[2], x3 = xr[3];
        feat[tid][0] = x0 * x0; feat[tid][1] = x1 * x1;
        feat[tid][2] = x2 * x2; feat[tid][3] = x3 * x3;
        feat[tid][4] = x0; feat[tid][5] = x1;
        feat[tid][6] = x2; feat[tid][7] = x3;
        feat[tid][8] = 1.f; feat[tid][9] = 0.f;
        feat[tid][10] = 0.f; feat[tid][11] = 0.f;
    }
    __syncthreads();

    int lane = tid & 31;
    int wv   = tid >> 5;             // wave index = N-tile (c block of 16)
    int mc   = lane & 15;
    int hi   = lane >> 4;            // 0: K+0/K+1 , 1: K+2/K+3  (ISA A/B layout)

    v8f acc = {};
#pragma unroll
    for (int kk = 0; kk < 3; ++kk) {
        v2f af, bf;
        int k0 = 4 * kk + 2 * hi;
        af.x = feat[mc][k0 + 0];
        af.y = feat[mc][k0 + 1];
        bf.x = Wl[k0 + 0][wv * 16 + mc];
        bf.y = Wl[k0 + 1][wv * 16 + mc];
        acc = __builtin_amdgcn_wmma_f32_16x16x4_f32(
            false, af, false, bf, (short)0, acc, false, false);
    }
    // D layout: reg r -> (M = r + 8*hi, N = mc)
    long rowbase = tile * 16;
#pragma unroll
    for (int r = 0; r < 8; ++r) {
        long row = rowbase + r + 8 * hi;
        logp[row * 64 + wv * 16 + mc] = acc[r];
    }
}

// ---------------------------------------------------------------------------
// K2: sequential forward + Viterbi scan. One block per sequence b.
//     128 threads: c = tid&63, h = tid>>6 splits the 64-term j-loop in two.
//     A/logA columns live in VGPRs; p_prev/d_prev broadcast from LDS.
// ---------------------------------------------------------------------------
__global__ void __launch_bounds__(128) hmm_scan(
    const float* __restrict__ x,
    const float* __restrict__ logp,     // nullptr -> compute emission inline
    const float* __restrict__ A,        // [64][64]
    const float* __restrict__ logA,     // [64][64]
    const float* __restrict__ logpi_g,  // [64]
    const float* __restrict__ W,        // [12][64]
    unsigned char* __restrict__ psi,    // [B][T-1][64]
    int* __restrict__ clast,            // [B]
    float* __restrict__ out_logpx)      // [B]
{
    __shared__ __align__(16) float p_prev[64];
    __shared__ __align__(16) float d_prev[64];
    __shared__ float sq[2][64], sm[2][64];
    __shared__ int   sa[2][64];
    __shared__ float redm[4], reds[4];

    const int tid  = threadIdx.x;
    const int c    = tid & 63;
    const int h    = tid >> 6;
    const int lane = tid & 31;
    const int wv   = tid >> 5;
    const int b    = blockIdx.x;

    // resident transition tables: my half of column c
    float Ac[32], lAc[32];
#pragma unroll
    for (int jj = 0; jj < 32; ++jj) {
        int j = h * 32 + jj;
        Ac[jj]  = A[j * 64 + c];
        lAc[jj] = logA[j * 64 + c];
    }
    // resident emission coefficients for state c
    float cA[4], cB[4];
#pragma unroll
    for (int d = 0; d < 4; ++d) {
        cA[d] = W[d * 64 + c];
        cB[d] = W[(4 + d) * 64 + c];
    }
    const float cCst = W[8 * 64 + c];
    const float lpi  = logpi_g[c];

    auto emis = [&](int t) -> float {
        if (logp) {
            long idx = ((long)b * TT + t) * 64 + c;
            if (t + 8 < TT) __builtin_prefetch(&logp[idx + 8 * 64], 0, 1);
            return logp[idx];
        }
        const float* xr = x + ((long)b * TT + t) * XD;
        float x0 = xr[0], x1 = xr[1], x2 = xr[2], x3 = xr[3];
        float v = cCst;
        v += (cA[0] * x0 + cB[0]) * x0;
        v += (cA[1] * x1 + cB[1]) * x1;
        v += (cA[2] * x2 + cB[2]) * x2;
        v += (cA[3] * x3 + cB[3]) * x3;
        return v;
    };
    auto blockMax = [&](float v) -> float {
#pragma unroll
        for (int o = 16; o > 0; o >>= 1) v = fmaxf(v, __shfl_xor(v, o, 32));
        if (lane == 0) redm[wv] = v;
        __syncthreads();
        return fmaxf(fmaxf(redm[0], redm[1]), fmaxf(redm[2], redm[3]));
    };
    auto blockSum = [&](float v) -> float {
#pragma unroll
        for (int o = 16; o > 0; o >>= 1) v += __shfl_xor(v, o, 32);
        if (lane == 0) reds[wv] = v;
        __syncthreads();
        return reds[0] + reds[1];   // h=0 waves cover all 64 states
    };

    // ---- t = 0 ----
    double s_acc;
    {
        float lp = emis(0);
        float a  = lp + lpi;
        float m  = blockMax(a);
        float E  = expf(a - m);
        float tot = blockSum(E);
        s_acc = (double)(m + logf(tot));
        if (h == 0) { p_prev[c] = E / tot; d_prev[c] = a; }
    }

    // ---- t = 1 .. T-1 ----
    for (int t = 1; t < TT; ++t) {
        __syncthreads();   // p_prev/d_prev visible
        float q = 0.f, best = -INFINITY;
        int   barg = 0;
        const float4* p4 = (const float4*)p_prev + h * 8;
        const float4* d4 = (const float4*)d_prev + h * 8;
#pragma unroll
        for (int u = 0; u < 8; ++u) {
            float4 pv = p4[u];
            float4 dv = d4[u];
            int j0 = h * 32 + u * 4;
            q = fmaf(pv.x, Ac[u * 4 + 0], q);
            q = fmaf(pv.y, Ac[u * 4 + 1], q);
            q = fmaf(pv.z, Ac[u * 4 + 2], q);
            q = fmaf(pv.w, Ac[u * 4 + 3], q);
            float s0 = dv.x + lAc[u * 4 + 0]; if (s0 > best) { best = s0; barg = j0 + 0; }
            float s1 = dv.y + lAc[u * 4 + 1]; if (s1 > best) { best = s1; barg = j0 + 1; }
            float s2 = dv.z + lAc[u * 4 + 2]; if (s2 > best) { best = s2; barg = j0 + 2; }
            float s3 = dv.w + lAc[u * 4 + 3]; if (s3 > best) { best = s3; barg = j0 + 3; }
        }
        sq[h][c] = q; sm[h][c] = best; sa[h][c] = barg;
        __syncthreads();
        float qf = sq[0][c] + sq[1][c];
        float b0 = sm[0][c], b1 = sm[1][c];
        int   arg = (b1 > b0) ? sa[1][c] : sa[0][c];   // ties -> smaller j
        float bestc = fmaxf(b0, b1);

        float lp = emis(t);
        float a  = lp + logf(qf);
        float m  = blockMax(a);
        float E  = expf(a - m);
        float tot = blockSum(E);
        s_acc += (double)(m + logf(tot));
        if (h == 0) {
            p_prev[c] = E / tot;
            d_prev[c] = bestc + lp;
            psi[((long)b * (TT - 1) + (t - 1)) * 64 + c] = (unsigned char)arg;
        }
    }

    // ---- argmax of final Viterbi scores ----
    __syncthreads();
    float dv = d_prev[c];
    int   di = c;
#pragma unroll
    for (int o = 16; o > 0; o >>= 1) {
        float ov = __shfl_xor(dv, o, 32);
        int   oi = __shfl_xor(di, o, 32);
        if (ov > dv || (ov == dv && oi < di)) { dv = ov; di = oi; }
    }
    if (lane == 0) { sm[0][wv] = dv; sa[0][wv] = di; }
    __syncthreads();
    if (tid == 0) {
        float v0 = sm[0][0]; int i0 = sa[0][0];
        float v1 = sm[0][1]; int i1 = sa[0][1];
        int ci = (v1 > v0 || (v1 == v0 && i1 < i0)) ? i1 : i0;
        clast[b] = ci;
        out_logpx[b] = (float)s_acc;
    }
}

// ---------------------------------------------------------------------------
// K3: Viterbi backtrack. One thread per sequence (128 parallel u8 chains in L2).
// ---------------------------------------------------------------------------
__global__ void hmm_backtrack(const unsigned char* __restrict__ psi,
                              const int* __restrict__ clast,
                              float* __restrict__ out_c)   // [B][T]
{
    int b = blockIdx.x * blockDim.x + threadIdx.x;
    if (b >= BB) return;
    int ci = clast[b];
    out_c[(long)b * TT + (TT - 1)] = (float)ci;
    for (int k = TT - 2; k >= 0; --k) {
        ci = psi[((long)b * (TT - 1) + k) * 64 + ci];
        out_c[(long)b * TT + k] = (float)ci;
    }
}

// ---------------------------------------------------------------------------
extern "C" void kernel_launch(void* const* d_in, const int* in_sizes, int n_in,
                              void* d_out, int out_size, void* d_ws, size_t ws_size,
                              hipStream_t stream)
{
    (void)in_sizes; (void)n_in; (void)out_size;
    const float* x      = (const float*)d_in[0];
    const float* means  = (const float*)d_in[1];
    const float* stds   = (const float*)d_in[2];
    const float* log_A  = (const float*)d_in[3];
    const float* log_pi = (const float*)d_in[4];
    float* out = (float*)d_out;

    char* ws = (char*)d_ws;
    size_t off = 0;
    auto wsalloc = [&](size_t bytes) -> void* {
        void* p = ws + off;
        off = (off + bytes + 255) & ~(size_t)255;
        return p;
    };
    float* A      = (float*)wsalloc(64 * 64 * sizeof(float));
    float* lA     = (float*)wsalloc(64 * 64 * sizeof(float));
    float* lpi    = (float*)wsalloc(64 * sizeof(float));
    float* W      = (float*)wsalloc(12 * 64 * sizeof(float));
    int*   clast  = (int*)wsalloc(BB * sizeof(int));
    unsigned char* psi = (unsigned char*)wsalloc((size_t)BB * (TT - 1) * 64);
    float* logp   = (float*)(ws + off);
    size_t need_full = off + (size_t)BB * TT * 64 * sizeof(float);
    bool use_logp = (ws_size >= need_full);

    hmm_prep<<<1, 64, 0, stream>>>(log_A, log_pi, means, stds, A, lA, lpi, W);
    if (use_logp) {
        hmm_emission_wmma<<<(BB * TT) / 16, 128, 0, stream>>>(x, W, logp);
    }
    hmm_scan<<<BB, 128, 0, stream>>>(x, use_logp ? logp : nullptr,
                                     A, lA, lpi, W, psi, clast, out);
    hmm_backtrack<<<1, 128, 0, stream>>>(psi, clast, out + BB);
}